// DagnabbitAutoEncoder_31164282700166
// MI455X (gfx1250) — compile-verified
//
#include <hip/hip_runtime.h>
#include <math.h>

// Problem constants (match reference setup_inputs)
#define D_    256           // embedding dim
#define KIN_  512           // IN_DEG * D (MLP input)
#define HID_  512           // 2*D (MLP hidden)
#define T_    4             // node types
#define R_    256           // root nodes
#define N_    65536         // trunk nodes
#define LMAX_ 4096          // max supported DAG depth (random DAG depth ~ O(log N) << this)
#define NKEY_ (LMAX_ * 4)   // sort keys = level*4 + type

typedef __attribute__((ext_vector_type(2))) float v2f;
typedef __attribute__((ext_vector_type(8))) float v8f;

// ---------------------------------------------------------------------------
// Branch-free erf, matching XLA's f32 ErfImpl (the lowering the JAX reference
// actually uses on GPU): erf(x) ~= x * P(x^2) / Q(x^2), x clamped to [-4, 4].
// 14 straight-line FMAs -> co-executes with the XDL/WMMA pipe, no EXEC
// divergence (unlike libm erff's |x|<1 branch which serializes both paths).
// ---------------------------------------------------------------------------
__device__ __forceinline__ float erf_xla(float y) {
  float x  = fminf(4.0f, fmaxf(-4.0f, y));
  float x2 = x * x;
  float a = -2.72614225801306e-10f;
  a = fmaf(a, x2,  2.77068142495902e-08f);
  a = fmaf(a, x2, -2.10102402082508e-06f);
  a = fmaf(a, x2, -5.69250639462346e-05f);
  a = fmaf(a, x2, -7.34990630326855e-04f);
  a = fmaf(a, x2, -2.95459980854025e-03f);
  a = fmaf(a, x2, -1.60960333262415e-02f);
  float b = -1.45660718464996e-05f;
  b = fmaf(b, x2, -2.13374055278905e-04f);
  b = fmaf(b, x2, -1.68282697438203e-03f);
  b = fmaf(b, x2, -7.37332916720468e-03f);
  b = fmaf(b, x2, -1.42647390514189e-02f);
  return (x * a) / b;
}

__device__ __forceinline__ float gelu_exact(float x) {
  return 0.5f * x * (1.0f + erf_xla(x * 0.70710678118654752f));
}

// ---------------------------------------------------------------------------
// Kernel 0: init. buf[0:R*D] = root; lvl = -1; counts = 0; flags: roots ready.
// ---------------------------------------------------------------------------
__global__ void dag_init_kernel(const float* __restrict__ root, float* __restrict__ buf,
                                int* __restrict__ lvl, int* __restrict__ counts,
                                int* __restrict__ flags) {
  int i = blockIdx.x * blockDim.x + threadIdx.x;
  if (i < R_ * D_)  buf[i] = root[i];
  if (i < N_)       lvl[i] = -1;
  if (i < NKEY_)    counts[i] = 0;
  if (i < R_ + N_)  flags[i] = (i < R_) ? 1 : 0;
}

// ---------------------------------------------------------------------------
// Kernel 0b: pack weights so each lane's two B values (rows kk, kk+1 of the
// same column) are adjacent -> one global_load_b64 per WMMA B operand.
// Wp[t][k/2][n][k&1] = W[t][k][n]
// ---------------------------------------------------------------------------
__global__ void dag_pack_kernel(const float* __restrict__ W, float* __restrict__ Wp,
                                int K, int Ncols) {
  int i = blockIdx.x * blockDim.x + threadIdx.x;
  if (i >= T_ * K * Ncols) return;
  int n = i % Ncols;
  int k = (i / Ncols) % K;
  int t = i / (Ncols * K);
  Wp[(((size_t)t * (K >> 1) + (k >> 1)) * Ncols + n) * 2 + (k & 1)] = W[i];
}

// ---------------------------------------------------------------------------
// Kernel 1: per-node DAG level via spin-wait (one thread per trunk node).
// A node only waits on lower-indexed nodes; finished lanes store *inside*
// the loop before being EXEC-masked off, so intra-wave deps resolve.
// ---------------------------------------------------------------------------
__global__ void dag_levels_kernel(const int* __restrict__ idx, int* lvl) {
  int i = blockIdx.x * blockDim.x + threadIdx.x;
  if (i >= N_) return;
  int p0 = idx[2 * i + 0];
  int p1 = idx[2 * i + 1];
  int l0 = (p0 < R_) ? 0 : -1;
  int l1 = (p1 < R_) ? 0 : -1;
  bool done = false;
  while (!done) {
    if (l0 < 0) l0 = __hip_atomic_load(&lvl[p0 - R_], __ATOMIC_RELAXED, __HIP_MEMORY_SCOPE_AGENT);
    if (l1 < 0) l1 = __hip_atomic_load(&lvl[p1 - R_], __ATOMIC_RELAXED, __HIP_MEMORY_SCOPE_AGENT);
    if (l0 >= 0 && l1 >= 0) {
      int lv = (l0 > l1 ? l0 : l1) + 1;
      __hip_atomic_store(&lvl[i], lv, __ATOMIC_RELEASE, __HIP_MEMORY_SCOPE_AGENT);
      done = true;
    } else {
      __builtin_amdgcn_s_sleep(1);
    }
  }
}

// ---------------------------------------------------------------------------
// Kernel 2: histogram of sort keys (level*4 + type).
// ---------------------------------------------------------------------------
__global__ void dag_hist_kernel(const int* __restrict__ lvl, const int* __restrict__ types,
                                int* counts) {
  int i = blockIdx.x * blockDim.x + threadIdx.x;
  if (i >= N_) return;
  int key = lvl[i] * 4 + types[i];
  if (key >= NKEY_) key = NKEY_ - 1;   // safety clamp (unreachable for these inputs)
  atomicAdd(&counts[key], 1);
}

// ---------------------------------------------------------------------------
// Kernel 3: single-block exclusive prefix sum over NKEY_ counts -> cursors.
// ---------------------------------------------------------------------------
__global__ void dag_scan_kernel(const int* __restrict__ counts, int* __restrict__ cursors) {
  __shared__ int partial[256];
  const int tid = threadIdx.x;
  const int CH = NKEY_ / 256;  // 64
  int s = 0;
  for (int j = 0; j < CH; ++j) s += counts[tid * CH + j];
  partial[tid] = s;
  __syncthreads();
  if (tid == 0) {
    int acc = 0;
    for (int j = 0; j < 256; ++j) { int v = partial[j]; partial[j] = acc; acc += v; }
  }
  __syncthreads();
  int acc = partial[tid];
  for (int j = 0; j < CH; ++j) {
    int v = counts[tid * CH + j];
    cursors[tid * CH + j] = acc;
    acc += v;
  }
}

// ---------------------------------------------------------------------------
// Kernel 4: counting-sort scatter: order[] = node ids sorted by key.
// ---------------------------------------------------------------------------
__global__ void dag_scatter_kernel(const int* __restrict__ lvl, const int* __restrict__ types,
                                   int* cursors, int* __restrict__ order) {
  int i = blockIdx.x * blockDim.x + threadIdx.x;
  if (i >= N_) return;
  int key = lvl[i] * 4 + types[i];
  if (key >= NKEY_) key = NKEY_ - 1;
  int pos = atomicAdd(&cursors[key], 1);
  order[pos] = i;
}

// ---------------------------------------------------------------------------
// Kernel 5: batched MLP over 16-node tiles with f32 WMMA.
// One block = 16 consecutive entries of the (level,type)-sorted order array.
// Runs of equal key inside a tile are processed sequentially (nearly always
// a single run). Parents are always in earlier buckets or earlier runs of
// this tile -> spin on ready flags, no deadlock (blocks dispatch ascending).
//
// WMMA f32 16x16x4 VGPR layouts (ISA 7.12.2):
//   A 16x4:  lanes 0-15 (M=lane):  v0=K0, v1=K1 ; lanes 16-31: v0=K2, v1=K3
//   B 4x16:  lanes 0-15 (N=lane):  v0=K0, v1=K1 ; lanes 16-31: v0=K2, v1=K3
//   C 16x16: vgpr r: lanes 0-15 -> (M=r, N=lane); lanes 16-31 -> (M=r+8, N=lane-16)
//
// Each wave carries NT accumulators (4 in GEMM1, 2 in GEMM2): the A fragment
// is loaded from LDS once per K-step and reused for all NT column tiles,
// giving denser back-to-back WMMA issue. The per-column bias is splatted into
// the accumulator initializer (every element of a C VGPR shares one column).
// ---------------------------------------------------------------------------
template <bool PACKED>
__global__ __launch_bounds__(256) void dag_compute_kernel(
    const float* __restrict__ W1, const float* __restrict__ b1,
    const float* __restrict__ W2, const float* __restrict__ b2,
    const float* __restrict__ W1p, const float* __restrict__ W2p,
    const int* __restrict__ idx, const int* __restrict__ types,
    const int* __restrict__ lvl, const int* __restrict__ order,
    int* flags, float* buf) {
  __shared__ __align__(16) float As[16][KIN_ + 4];  // +4 pad: LDS bank spread
  __shared__ __align__(16) float Hs[16][HID_ + 4];
  __shared__ int nids[16], tys[16], kys[16], par0[16], par1[16];

  const int tid = threadIdx.x;
  const int tile = blockIdx.x;

  if (tid < 16) {
    int nid = order[tile * 16 + tid];
    nids[tid] = nid;
    tys[tid]  = types[nid];
    kys[tid]  = lvl[nid] * 4 + types[nid];
    par0[tid] = idx[2 * nid + 0];
    par1[tid] = idx[2 * nid + 1];
  }
  __syncthreads();

  const int lane = tid & 31;
  const int wv   = tid >> 5;      // 8 wave32s
  const int half = lane >> 4;     // 0: lanes 0-15, 1: lanes 16-31
  const int m    = lane & 15;

  int r0 = 0;
  while (r0 < 16) {
    // --- find run [r0, r1) of equal (level,type) key (uniform across block) ---
    int key0 = kys[r0];
    int r1 = r0 + 1;
    while (r1 < 16 && kys[r1] == key0) ++r1;
    const int t = tys[r0];

    // --- wait until parents of this run are published ---
    if (tid >= r0 && tid < r1) {
      int p0 = par0[tid], p1 = par1[tid];
      if (p0 >= R_)
        while (__hip_atomic_load(&flags[p0], __ATOMIC_ACQUIRE, __HIP_MEMORY_SCOPE_AGENT) == 0)
          __builtin_amdgcn_s_sleep(1);
      if (p1 >= R_)
        while (__hip_atomic_load(&flags[p1], __ATOMIC_ACQUIRE, __HIP_MEMORY_SCOPE_AGENT) == 0)
          __builtin_amdgcn_s_sleep(1);
    }
    __syncthreads();

    // --- gather A rows for this run: x = [buf[p0], buf[p1]] (float4 vectorized) ---
    const int rows = r1 - r0;
    for (int w = tid; w < rows * 128; w += 256) {
      int row = r0 + (w >> 7);
      int j   = w & 127;                      // 128 float4 per row
      int p   = (j < 64) ? par0[row] : par1[row];
      int c   = (j & 63) << 2;                // col within parent
      float4 s = *(const float4*)&buf[(size_t)p * D_ + c];
      *(float4*)&As[row][(j < 64) ? c : (D_ + c)] = s;
    }
    __syncthreads();

    // --- GEMM1: H = gelu(A(16x512) @ W1[t](512x512) + b1[t]) ---
    {
      const float* Wt  = PACKED ? (W1p + (size_t)t * (KIN_ / 2) * HID_ * 2)
                                : (W1  + (size_t)t * KIN_ * HID_);
      const float* b1t = b1 + (size_t)t * HID_;
      const int n0 = (wv << 6) + m;           // wave owns 64 contiguous columns
      v8f acc[4];
      #pragma unroll
      for (int j = 0; j < 4; ++j) {           // splat per-column bias into C
        float bias = b1t[n0 + (j << 4)];
        #pragma unroll
        for (int r = 0; r < 8; ++r) acc[j][r] = bias;
      }
      for (int k0 = 0; k0 < KIN_; k0 += 4) {  // 128 K-steps, A loaded once/step
        const int kk = k0 + 2 * half;
        v2f a; a.x = As[m][kk]; a.y = As[m][kk + 1];
        #pragma unroll
        for (int j = 0; j < 4; ++j) {
          v2f b;
          if (PACKED) {
            b = *(const v2f*)(Wt + ((size_t)(kk >> 1) * HID_ + n0 + (j << 4)) * 2);
          } else {
            const float* Wp = Wt + n0 + (j << 4);
            b.x = Wp[(size_t)kk * HID_];
            b.y = Wp[(size_t)(kk + 1) * HID_];
          }
          acc[j] = __builtin_amdgcn_wmma_f32_16x16x4_f32(false, a, false, b,
                                                         (short)0, acc[j], false, false);
        }
      }
      #pragma unroll
      for (int j = 0; j < 4; ++j)
        #pragma unroll
        for (int r = 0; r < 8; ++r)
          Hs[r + 8 * half][n0 + (j << 4)] = gelu_exact(acc[j][r]);
    }
    __syncthreads();

    // --- GEMM2: E = H(16x512) @ W2[t](512x256) + b2[t]; scatter run rows ---
    {
      const float* Wt  = PACKED ? (W2p + (size_t)t * (HID_ / 2) * D_ * 2)
                                : (W2  + (size_t)t * HID_ * D_);
      const float* b2t = b2 + (size_t)t * D_;
      const int n0 = (wv << 5) + m;           // wave owns 32 contiguous columns
      v8f acc[2];
      #pragma unroll
      for (int j = 0; j < 2; ++j) {
        float bias = b2t[n0 + (j << 4)];
        #pragma unroll
        for (int r = 0; r < 8; ++r) acc[j][r] = bias;
      }
      for (int k0 = 0; k0 < HID_; k0 += 4) {
        const int kk = k0 + 2 * half;
        v2f a; a.x = Hs[m][kk]; a.y = Hs[m][kk + 1];
        #pragma unroll
        for (int j = 0; j < 2; ++j) {
          v2f b;
          if (PACKED) {
            b = *(const v2f*)(Wt + ((size_t)(kk >> 1) * D_ + n0 + (j << 4)) * 2);
          } else {
            const float* Wp = Wt + n0 + (j << 4);
            b.x = Wp[(size_t)kk * D_];
            b.y = Wp[(size_t)(kk + 1) * D_];
          }
          acc[j] = __builtin_amdgcn_wmma_f32_16x16x4_f32(false, a, false, b,
                                                         (short)0, acc[j], false, false);
        }
      }
      #pragma unroll
      for (int j = 0; j < 2; ++j)
        #pragma unroll
        for (int r = 0; r < 8; ++r) {
          const int row = r + 8 * half;
          if (row >= r0 && row < r1)
            buf[(size_t)(R_ + nids[row]) * D_ + n0 + (j << 4)] = acc[j][r];
        }
    }

    // --- publish: device fence, then release flags for this run's nodes ---
    __threadfence();
    __syncthreads();
    if (tid >= r0 && tid < r1)
      __hip_atomic_store(&flags[R_ + nids[tid]], 1, __ATOMIC_RELEASE, __HIP_MEMORY_SCOPE_AGENT);

    r0 = r1;
  }
}

// ---------------------------------------------------------------------------
// Host launcher
// ---------------------------------------------------------------------------
extern "C" void kernel_launch(void* const* d_in, const int* in_sizes, int n_in,
                              void* d_out, int out_size, void* d_ws, size_t ws_size,
                              hipStream_t stream) {
  const float* root = (const float*)d_in[0];   // (R, D)
  const float* W1   = (const float*)d_in[1];   // (T, 512, 512)
  const float* b1   = (const float*)d_in[2];   // (T, 512)
  const float* W2   = (const float*)d_in[3];   // (T, 512, 256)
  const float* b2   = (const float*)d_in[4];   // (T, 256)
  const int*   idx  = (const int*)d_in[5];     // (N, 2)
  const int*   ty   = (const int*)d_in[6];     // (N,)
  float* buf = (float*)d_out;                  // (R+N, D) — doubles as shared embedding buffer
  (void)in_sizes; (void)n_in; (void)out_size;

  const size_t w1Elems = (size_t)T_ * KIN_ * HID_;   // 1M floats (4 MB)
  const size_t w2Elems = (size_t)T_ * HID_ * D_;     // 512K floats (2 MB)
  const size_t metaInts = (size_t)N_ + 2 * NKEY_ + N_ + (R_ + N_);
  const size_t packedBytes = (w1Elems + w2Elems) * sizeof(float) + metaInts * sizeof(int);
  const bool usePacked = ws_size >= packedBytes;     // constant across calls -> deterministic

  float* W1p = (float*)d_ws;                   // packed weights first (16B-aligned base)
  float* W2p = W1p + w1Elems;
  int* meta  = usePacked ? (int*)(W2p + w2Elems) : (int*)d_ws;
  int* lvl     = meta;                 // N
  int* counts  = lvl + N_;             // NKEY
  int* cursors = counts + NKEY_;       // NKEY
  int* order   = cursors + NKEY_;      // N
  int* flags   = order + N_;           // R+N

  dag_init_kernel<<<(R_ + N_ + 255) / 256, 256, 0, stream>>>(root, buf, lvl, counts, flags);
  if (usePacked) {
    dag_pack_kernel<<<(int)((w1Elems + 255) / 256), 256, 0, stream>>>(W1, W1p, KIN_, HID_);
    dag_pack_kernel<<<(int)((w2Elems + 255) / 256), 256, 0, stream>>>(W2, W2p, HID_, D_);
  }
  dag_levels_kernel<<<N_ / 256, 256, 0, stream>>>(idx, lvl);
  dag_hist_kernel<<<N_ / 256, 256, 0, stream>>>(lvl, ty, counts);
  dag_scan_kernel<<<1, 256, 0, stream>>>(counts, cursors);
  dag_scatter_kernel<<<N_ / 256, 256, 0, stream>>>(lvl, ty, cursors, order);
  if (usePacked) {
    dag_compute_kernel<true><<<N_ / 16, 256, 0, stream>>>(W1, b1, W2, b2, W1p, W2p,
                                                          idx, ty, lvl, order, flags, buf);
  } else {
    dag_compute_kernel<false><<<N_ / 16, 256, 0, stream>>>(W1, b1, W2, b2, W1p, W2p,
                                                           idx, ty, lvl, order, flags, buf);
  }
}